// GCN_6811818131825
// MI455X (gfx1250) — compile-verified
//
#include <hip/hip_runtime.h>
#include <math.h>

typedef __attribute__((ext_vector_type(2))) float v2f;
typedef __attribute__((ext_vector_type(8))) float v8f;

#define NFEAT  256
#define NHID   128
#define NCLASS 40

// ---------------------------------------------------------------------------
// dst[i*cols + c] = bias[c]   (row-broadcast init for the accumulation buffers)
// ---------------------------------------------------------------------------
__global__ void init_bias_kernel(float* __restrict__ dst, const float* __restrict__ bias,
                                 long long total, int cols) {
    long long t = (long long)blockIdx.x * blockDim.x + threadIdx.x;
    if (t < total) {
        int c = (int)(t % cols);
        dst[t] = bias[c];
    }
}

// ---------------------------------------------------------------------------
// Y[nRows x 128] = X[nRows x 256] @ W[256 x 128]  via V_WMMA_F32_16X16X4_F32.
// 8 waves per block; wave w owns output tile (blockIdx.x*16, w*16).
// A frag (16x4 f32): lanes 0-15 hold K=k,k+1 ; lanes 16-31 hold K=k+2,k+3.
// B frag (4x16 f32): mirrored layout (row K striped across lanes in a VGPR).
// D (16x16 f32, 8 VGPRs): VGPR r -> M = r (lanes 0-15) / M = r+8 (lanes 16-31).
// Tile-uniform bounds check: single scalar branch, unguarded stores fast path.
// ---------------------------------------------------------------------------
__global__ void gemm_xw0_kernel(const float* __restrict__ X, const float* __restrict__ W,
                                float* __restrict__ Y, int nRows) {
    const int lane  = threadIdx.x & 31;
    const int wave  = threadIdx.x >> 5;          // 0..7 -> N tile of NHID
    const int mBase = blockIdx.x * 16;
    const int nBase = wave * 16;
    const int lm    = lane & 15;
    const int koff  = (lane >> 4) << 1;          // 0 or 2
    const bool fullTile = (mBase + 16 <= nRows); // wave-uniform

    int mrow = mBase + lm;
    if (mrow >= nRows) mrow = nRows - 1;         // clamp (edge tile only)
    const float* Arow = X + (size_t)mrow * NFEAT;

    v8f acc = {0.f, 0.f, 0.f, 0.f, 0.f, 0.f, 0.f, 0.f};
    for (int k = 0; k < NFEAT; k += 4) {
        v2f a, b;
        a.x = Arow[k + koff];
        a.y = Arow[k + koff + 1];
        b.x = W[(size_t)(k + koff) * NHID + nBase + lm];
        b.y = W[(size_t)(k + koff + 1) * NHID + nBase + lm];
        acc = __builtin_amdgcn_wmma_f32_16x16x4_f32(false, a, false, b,
                                                    (short)0, acc, false, false);
    }

    const int rowBase = mBase + ((lane >> 4) << 3);
    float* yp = Y + (size_t)rowBase * NHID + nBase + lm;
    if (fullTile) {
#pragma unroll
        for (int r = 0; r < 8; ++r) yp[(size_t)r * NHID] = acc[r];
    } else {
#pragma unroll
        for (int r = 0; r < 8; ++r)
            if (rowBase + r < nRows) yp[(size_t)r * NHID] = acc[r];
    }
}

// ---------------------------------------------------------------------------
// Y[nRows x 40] = H[nRows x 128] @ W[128 x 40], N padded to 48 (3 waves/block).
// ---------------------------------------------------------------------------
__global__ void gemm_hw1_kernel(const float* __restrict__ H, const float* __restrict__ W,
                                float* __restrict__ Y, int nRows) {
    const int lane  = threadIdx.x & 31;
    const int wave  = threadIdx.x >> 5;          // 0..2
    const int mBase = blockIdx.x * 16;
    const int nBase = wave * 16;
    const int lm    = lane & 15;
    const int koff  = (lane >> 4) << 1;
    const int col   = nBase + lm;
    const bool colOK    = (col < NCLASS);
    const bool fullTile = (mBase + 16 <= nRows);

    int mrow = mBase + lm;
    if (mrow >= nRows) mrow = nRows - 1;
    const float* Arow = H + (size_t)mrow * NHID;

    v8f acc = {0.f, 0.f, 0.f, 0.f, 0.f, 0.f, 0.f, 0.f};
    for (int k = 0; k < NHID; k += 4) {
        v2f a, b;
        a.x = Arow[k + koff];
        a.y = Arow[k + koff + 1];
        b.x = colOK ? W[(size_t)(k + koff) * NCLASS + col] : 0.f;
        b.y = colOK ? W[(size_t)(k + koff + 1) * NCLASS + col] : 0.f;
        acc = __builtin_amdgcn_wmma_f32_16x16x4_f32(false, a, false, b,
                                                    (short)0, acc, false, false);
    }

    if (colOK) {
        const int rowBase = mBase + ((lane >> 4) << 3);
        float* yp = Y + (size_t)rowBase * NCLASS + col;
        if (fullTile) {
#pragma unroll
            for (int r = 0; r < 8; ++r) yp[(size_t)r * NCLASS] = acc[r];
        } else {
#pragma unroll
            for (int r = 0; r < 8; ++r)
                if (rowBase + r < nRows) yp[(size_t)r * NCLASS] = acc[r];
        }
    }
}

// ---------------------------------------------------------------------------
// H[dst[e]] += w[e] * XW[src[e]]   (128 feats; one wave per edge, float4/lane)
// ---------------------------------------------------------------------------
__global__ void spmm_h_kernel(const float* __restrict__ XW, const int* __restrict__ src,
                              const int* __restrict__ dst, const float* __restrict__ w,
                              float* __restrict__ H, int nEdges) {
    const int lane = threadIdx.x & 31;
    const int e    = blockIdx.x * 8 + (threadIdx.x >> 5);
    if (e >= nEdges) return;
    const float we = w[e];
    const int   s  = src[e];
    const int   d  = dst[e];
    const float4 v = *(const float4*)(XW + (size_t)s * NHID + lane * 4);
    float* out = H + (size_t)d * NHID + lane * 4;
    atomicAdd(out + 0, we * v.x);
    atomicAdd(out + 1, we * v.y);
    atomicAdd(out + 2, we * v.z);
    atomicAdd(out + 3, we * v.w);
}

// ---------------------------------------------------------------------------
// Z[dst[e]][c] += w[e] * HW[src[e]][c]   (40 classes; one thread per (e,c))
// ---------------------------------------------------------------------------
__global__ void spmm_z_kernel(const float* __restrict__ HW, const int* __restrict__ src,
                              const int* __restrict__ dst, const float* __restrict__ w,
                              float* __restrict__ Z, int nEdges) {
    long long t     = (long long)blockIdx.x * blockDim.x + threadIdx.x;
    long long total = (long long)nEdges * NCLASS;
    if (t >= total) return;
    int e = (int)(t / NCLASS);
    int c = (int)(t % NCLASS);
    atomicAdd(&Z[(size_t)dst[e] * NCLASS + c], w[e] * HW[(size_t)src[e] * NCLASS + c]);
}

// ---------------------------------------------------------------------------
// Row-wise: h = relu(h / max(||h||_2, eps)).  One wave32 per node, float4/lane.
// ---------------------------------------------------------------------------
__global__ void norm_relu_kernel(float* __restrict__ H, int nRows) {
    const int lane = threadIdx.x & 31;
    const int node = blockIdx.x * 8 + (threadIdx.x >> 5);
    if (node >= nRows) return;
    float* p = H + (size_t)node * NHID + lane * 4;
    float4 v = *(float4*)p;
    float ss = v.x * v.x + v.y * v.y + v.z * v.z + v.w * v.w;
#pragma unroll
    for (int m = 16; m >= 1; m >>= 1) ss += __shfl_xor(ss, m, 32);
    const float inv = 1.0f / fmaxf(sqrtf(ss), 1e-12f);
    v.x = fmaxf(v.x * inv, 0.f);
    v.y = fmaxf(v.y * inv, 0.f);
    v.z = fmaxf(v.z * inv, 0.f);
    v.w = fmaxf(v.w * inv, 0.f);
    *(float4*)p = v;
}

// ---------------------------------------------------------------------------
// In-place log_softmax over 40 classes. One wave32 per node; lane covers
// columns {lane, lane+32}.
// ---------------------------------------------------------------------------
__global__ void log_softmax_kernel(float* __restrict__ Z, int nRows) {
    const int lane = threadIdx.x & 31;
    const int node = blockIdx.x * 8 + (threadIdx.x >> 5);
    if (node >= nRows) return;
    float* row = Z + (size_t)node * NCLASS;
    const int  c2   = lane + 32;
    const bool has2 = (c2 < NCLASS);
    float z1 = row[lane];                          // lane < 32 < 40: always valid
    float z2 = has2 ? row[c2] : -INFINITY;

    float m = fmaxf(z1, z2);
#pragma unroll
    for (int s = 16; s >= 1; s >>= 1) m = fmaxf(m, __shfl_xor(m, s, 32));
    float es = __expf(z1 - m) + (has2 ? __expf(z2 - m) : 0.f);
#pragma unroll
    for (int s = 16; s >= 1; s >>= 1) es += __shfl_xor(es, s, 32);
    const float lse = m + __logf(es);
    row[lane] = z1 - lse;
    if (has2) row[c2] = z2 - lse;
}

// ---------------------------------------------------------------------------
extern "C" void kernel_launch(void* const* d_in, const int* in_sizes, int n_in,
                              void* d_out, int out_size, void* d_ws, size_t ws_size,
                              hipStream_t stream) {
    const float* x    = (const float*)d_in[0];
    const int*   esrc = (const int*)d_in[1];
    const int*   edst = (const int*)d_in[2];
    const float* ew   = (const float*)d_in[3];
    const float* W0   = (const float*)d_in[4];
    const float* b0   = (const float*)d_in[5];
    const float* W1   = (const float*)d_in[6];
    const float* b1   = (const float*)d_in[7];
    float* out = (float*)d_out;

    const int nNodes = in_sizes[0] / NFEAT;
    const int nEdges = in_sizes[1];

    // workspace: XW0 (later reused for HW1) | H1
    float* XW0 = (float*)d_ws;                     // nNodes*NHID floats
    float* H1  = XW0 + (size_t)nNodes * NHID;      // nNodes*NHID floats
    float* HW1 = XW0;                              // reuse after SpMM1 consumed XW0

    const int mTiles = (nNodes + 15) / 16;

    // 1) H1 = broadcast(b0)
    {
        long long total = (long long)nNodes * NHID;
        init_bias_kernel<<<(int)((total + 255) / 256), 256, 0, stream>>>(H1, b0, total, NHID);
    }
    // 2) XW0 = x @ W0  (WMMA f32)
    gemm_xw0_kernel<<<mTiles, 256, 0, stream>>>(x, W0, XW0, nNodes);

    // 3) H1 += scatter(w * XW0[src] -> dst)
    spmm_h_kernel<<<(nEdges + 7) / 8, 256, 0, stream>>>(XW0, esrc, edst, ew, H1, nEdges);

    // 4) H1 = relu(l2norm(H1))
    norm_relu_kernel<<<(nNodes + 7) / 8, 256, 0, stream>>>(H1, nNodes);

    // 5) HW1 = H1 @ W1  (WMMA f32, into recycled XW0 buffer)
    gemm_hw1_kernel<<<mTiles, 96, 0, stream>>>(H1, W1, HW1, nNodes);

    // 6) out = broadcast(b1)
    {
        long long total = (long long)nNodes * NCLASS;
        init_bias_kernel<<<(int)((total + 255) / 256), 256, 0, stream>>>(out, b1, total, NCLASS);
    }
    // 7) out += scatter(w * HW1[src] -> dst)
    {
        long long total = (long long)nEdges * NCLASS;
        spmm_z_kernel<<<(int)((total + 255) / 256), 256, 0, stream>>>(HW1, esrc, edst, ew, out, nEdges);
    }
    // 8) out = log_softmax(out) row-wise, in place
    log_softmax_kernel<<<(nNodes + 7) / 8, 256, 0, stream>>>(out, nNodes);
}